// TestHelionAllReduceAddRMSNormModel_17695265259656
// MI455X (gfx1250) — compile-verified
//
#include <hip/hip_runtime.h>

// ---------------------------------------------------------------------------
// Fused AllReduce(sum over 4 ranks) + residual add + RMSNorm for MI455X.
//
// Shapes: input_shared [4, 8192, 4096] f32, residual [8192, 4096] f32,
//         gamma [4096] f32.  Outputs (concatenated in d_out):
//         normed [8192,4096] f32, new_residual [8192,4096] f32.
//
// Memory-bound: ~0.94 GB total traffic -> ~40 us floor at 23.3 TB/s.
// Strategy: one 256-thread block (8 wave32) per token row.
//   * Stage the 4 rank rows (4 x 16 KB) into LDS with gfx1250 async DMA
//     (GLOBAL_LOAD_ASYNC_TO_LDS_B128, ASYNCcnt) -> 64 KB LDS, 5 blocks/WGP.
//   * Residual streamed with non-temporal b128 loads; gamma cached (RT).
//   * Accumulate 4 ranks + residual from LDS, NT-store new_residual,
//     wave32 shfl_xor reduction + cross-wave LDS combine for sum(x^2),
//     v_rsq, scale, NT-store normed.
// ---------------------------------------------------------------------------

#define TOKENS   8192
#define HIDDEN   4096
#define WORLD    4
#define BLOCK    256
#define NCHUNK   (HIDDEN / (BLOCK * 4))   // 4 float4 per thread
#define ROW_F4   (HIDDEN / 4)             // 1024 float4 per row
#define EPS      1e-6f

typedef float v4f __attribute__((ext_vector_type(4)));
typedef int   v4i __attribute__((ext_vector_type(4)));

// Builtin signature is (int4 AS1*, int4 AS3*, imm offset, imm cpol).
typedef __attribute__((address_space(1))) v4i* gas_v4i;   // global int4*
typedef __attribute__((address_space(3))) v4i* las_v4i;   // LDS int4*

__device__ __forceinline__ v4f nt_load4(const v4f* p) {
  return __builtin_nontemporal_load(p);
}
__device__ __forceinline__ void nt_store4(v4f* p, v4f v) {
  __builtin_nontemporal_store(v, p);
}

__global__ __launch_bounds__(BLOCK) void fused_allreduce_add_rmsnorm(
    const float* __restrict__ in,      // [WORLD, TOKENS, HIDDEN]
    const float* __restrict__ resid,   // [TOKENS, HIDDEN]
    const float* __restrict__ gamma,   // [HIDDEN]
    float*       __restrict__ out_normed,  // [TOKENS, HIDDEN]
    float*       __restrict__ out_resid)   // [TOKENS, HIDDEN]
{
  __shared__ v4f lds[WORLD * ROW_F4];      // 64 KB staging (reused as scratch)

  const int row = blockIdx.x;
  const int tid = threadIdx.x;

  // ---- 1) Async-DMA the four rank rows into LDS (ASYNCcnt path) ----------
#if __has_builtin(__builtin_amdgcn_global_load_async_to_lds_b128)
  #pragma unroll
  for (int r = 0; r < WORLD; ++r) {
    const v4f* g = (const v4f*)(in + ((size_t)r * TOKENS + row) * HIDDEN);
    #pragma unroll
    for (int c = 0; c < NCHUNK; ++c) {
      const int idx = c * BLOCK + tid;
      __builtin_amdgcn_global_load_async_to_lds_b128(
          (gas_v4i)(g + idx),
          (las_v4i)&lds[r * ROW_F4 + idx],
          /*offset=*/0, /*cpol=*/0);
    }
  }
#else
  // Fallback: plain global->register->LDS copy (still b128 + ds_store_b128).
  #pragma unroll
  for (int r = 0; r < WORLD; ++r) {
    const v4f* g = (const v4f*)(in + ((size_t)r * TOKENS + row) * HIDDEN);
    #pragma unroll
    for (int c = 0; c < NCHUNK; ++c) {
      const int idx = c * BLOCK + tid;
      lds[r * ROW_F4 + idx] = nt_load4(g + idx);
    }
  }
#endif

  // ---- 2) Overlap: residual (streaming, NT) + gamma (reused, RT) ---------
  const v4f* resid4 = (const v4f*)(resid + (size_t)row * HIDDEN);
  const v4f* gamma4 = (const v4f*)gamma;
  v4f resv[NCHUNK], gv[NCHUNK];
  #pragma unroll
  for (int c = 0; c < NCHUNK; ++c) {
    const int idx = c * BLOCK + tid;
    resv[c] = nt_load4(resid4 + idx);
    gv[c]   = gamma4[idx];
  }

  // ---- 3) Wait for this wave's async DMA, then sync the block ------------
#if __has_builtin(__builtin_amdgcn_s_wait_asynccnt)
  __builtin_amdgcn_s_wait_asynccnt(0);
#else
  asm volatile("s_wait_asynccnt 0" ::: "memory");
#endif
  __syncthreads();

  // ---- 4) Reduce 4 ranks + residual, emit new_residual, acc sum(x^2) -----
  v4f*  out_r4 = (v4f*)(out_resid  + (size_t)row * HIDDEN);
  v4f*  out_n4 = (v4f*)(out_normed + (size_t)row * HIDDEN);
  v4f   nr[NCHUNK];
  float ss = 0.0f;
  #pragma unroll
  for (int c = 0; c < NCHUNK; ++c) {
    const int idx = c * BLOCK + tid;
    v4f a = lds[0 * ROW_F4 + idx];
    a += lds[1 * ROW_F4 + idx];
    a += lds[2 * ROW_F4 + idx];
    a += lds[3 * ROW_F4 + idx];
    a += resv[c];
    nr[c] = a;
    nt_store4(out_r4 + idx, a);
    ss += a.x * a.x + a.y * a.y + a.z * a.z + a.w * a.w;
  }

  // ---- 5) wave32 reduction + cross-wave combine --------------------------
  #pragma unroll
  for (int off = 16; off > 0; off >>= 1)
    ss += __shfl_xor(ss, off, 32);

  __syncthreads();                       // all LDS reads done -> safe to reuse
  float* scratch = (float*)lds;
  const int wave = tid >> 5;
  if ((tid & 31) == 0) scratch[wave] = ss;
  __syncthreads();
  if (tid == 0) {
    float tot = 0.0f;
    #pragma unroll
    for (int w = 0; w < BLOCK / 32; ++w) tot += scratch[w];
    scratch[BLOCK / 32] = rsqrtf(tot * (1.0f / HIDDEN) + EPS);
  }
  __syncthreads();
  const float scale = scratch[BLOCK / 32];

  // ---- 6) normed = new_residual * rsqrt(mean(x^2)+eps) * gamma -----------
  #pragma unroll
  for (int c = 0; c < NCHUNK; ++c) {
    const int idx = c * BLOCK + tid;
    v4f o = nr[c];
    o.x *= scale * gv[c].x;
    o.y *= scale * gv[c].y;
    o.z *= scale * gv[c].z;
    o.w *= scale * gv[c].w;
    nt_store4(out_n4 + idx, o);
  }
}

extern "C" void kernel_launch(void* const* d_in, const int* in_sizes, int n_in,
                              void* d_out, int out_size, void* d_ws, size_t ws_size,
                              hipStream_t stream) {
  (void)in_sizes; (void)n_in; (void)out_size; (void)d_ws; (void)ws_size;
  const float* in    = (const float*)d_in[0];   // [4, 8192, 4096]
  const float* resid = (const float*)d_in[1];   // [8192, 4096]
  const float* gamma = (const float*)d_in[2];   // [4096]
  float* out_normed = (float*)d_out;                              // first output
  float* out_resid  = (float*)d_out + (size_t)TOKENS * HIDDEN;    // second output

  fused_allreduce_add_rmsnorm<<<dim3(TOKENS), dim3(BLOCK), 0, stream>>>(
      in, resid, gamma, out_normed, out_resid);
}